// MorphologicalModule_13237089207002
// MI455X (gfx1250) — compile-verified
//
#include <hip/hip_runtime.h>
#include <hip/hip_bf16.h>

typedef __attribute__((ext_vector_type(2))) float v2f;
typedef __attribute__((ext_vector_type(8))) float v8f;

// ---------------------------------------------------------------------------
// WMMA fp32 GEMM: out[m,n] = act( sum_k A[m,k] * W[n,k] + bias[n] )
// A: M x K row-major, W: N x K row-major (i.e., computes A @ W^T)
// One wave (32 threads) per block computes one 16x16 tile via
// V_WMMA_F32_16X16X4_F32, stepping K by 4.
//
// VGPR layouts per CDNA5 ISA 7.12.2:
//  A (16x4, MxK): lanes 0-15 hold K=0 (v0), K=1 (v1); lanes 16-31 K=2, K=3.
//  B (4x16, KxN): lanes 0-15 hold row K=0 (v0) / K=1 (v1); lanes 16-31 K=2/K=3.
//  C/D (16x16):   VGPR r, lanes 0-15 -> M=r; lanes 16-31 -> M=r+8; N=lane%16.
// ---------------------------------------------------------------------------
__global__ void gemm_wmma_relu(const float* __restrict__ A,
                               const float* __restrict__ W,
                               const float* __restrict__ bias,
                               float* __restrict__ out,
                               int M, int N, int K, int do_relu) {
  const int mt   = blockIdx.x;
  const int nt   = blockIdx.y;
  const int lane = threadIdx.x;      // 0..31 (wave32)
  const int half = lane >> 4;        // 0: K pair {0,1}; 1: K pair {2,3}
  const int l    = lane & 15;
  const int row0 = mt * 16;
  const int col0 = nt * 16;

  // Clamp B-row index so lanes beyond N (layer-2, N=10) never read OOB;
  // their products land in columns we never store.
  const int wrow = (col0 + l < N) ? (col0 + l) : (N - 1);

  v8f c = {};
  for (int k = 0; k < K; k += 4) {
    const int ka = k + half * 2;
    v2f a, b;
    a.x = A[(size_t)(row0 + l) * K + ka];
    a.y = A[(size_t)(row0 + l) * K + ka + 1];
    b.x = W[(size_t)wrow * K + ka];
    b.y = W[(size_t)wrow * K + ka + 1];
    // D = A*B + C  (fp32 WMMA, 8 args: neg_a, A, neg_b, B, c_mod, C, reuse_a, reuse_b)
    c = __builtin_amdgcn_wmma_f32_16x16x4_f32(false, a, false, b, (short)0, c,
                                              false, false);
  }

  const int col = col0 + l;
  if (col < N) {
    const float bn = bias[col];
#pragma unroll
    for (int r = 0; r < 8; ++r) {
      const int row = row0 + r + half * 8;
      float v = c[r] + bn;
      if (do_relu) v = fmaxf(v, 0.0f);
      out[(size_t)row * N + col] = v;
    }
  }
}

// ---------------------------------------------------------------------------
// beta (64x10) -> alpha (64), kern = sigmoid(beta[:,1:] + identity_bias) (64x9)
// identity_bias: element 0 -> 0, elements 1..9 -> -5 except center (flat idx 4)
// which is +5.
// ---------------------------------------------------------------------------
__global__ void mlp_epilogue(const float* __restrict__ beta,
                             float* __restrict__ alpha,
                             float* __restrict__ kern) {
  const int b = threadIdx.x;   // 0..63
  alpha[b] = beta[b * 10 + 0];
#pragma unroll
  for (int j = 0; j < 9; ++j) {
    float t = beta[b * 10 + 1 + j] + ((j == 4) ? 5.0f : -5.0f);
    kern[b * 9 + j] = 1.0f / (1.0f + __expf(-t));
  }
}

// ---------------------------------------------------------------------------
// Smoothmax: out[b,h,w] = sum_i p_i*e_i / sum_i e_i,
//   e_i = exp(alpha_b * p_i) * kern_b[i] over 3x3 zero-padded patch.
//
// Block = 8 output rows x 256 cols of one image (256 threads).
// 10 halo rows staged in LDS (10x258 floats ~ 10 KB of the 320 KB WGP LDS).
// Each thread sweeps its column downward with a rolling 3x3 register window:
// exp(alpha*p) is computed ONCE per input element and reused by the 3 output
// rows it feeds -> 3 exps + 3 LDS reads per pixel instead of 9 + 9.
// Halo overhead 10/8 -> ~21 MB read + 16.8 MB write, near the HBM floor.
// ---------------------------------------------------------------------------
__global__ void __launch_bounds__(256)
smoothmax_kernel(const float* __restrict__ x,
                 const float* __restrict__ alpha,
                 const float* __restrict__ kern,
                 float* __restrict__ out) {
  constexpr int H = 256, W = 256, TH = 8;     // TH output rows per block
  const int w  = threadIdx.x;                 // 0..255 (column)
  const int h0 = blockIdx.x * TH;             // first output row of tile
  const int b  = blockIdx.y;                  // image index

  __shared__ float tile[TH + 2][W + 2];       // rows h0-1 .. h0+TH, 1-col halo
  __shared__ float sk[9];
  __shared__ float sa;

  const float* img = x + (size_t)b * H * W;
#pragma unroll
  for (int r = 0; r < TH + 2; ++r) {
    const int hr = h0 - 1 + r;
    tile[r][w + 1] = (hr >= 0 && hr < H) ? img[(size_t)hr * W + w] : 0.0f;
  }
  if (w < TH + 2) {                           // zero column halos
    tile[w][0]     = 0.0f;
    tile[w][W + 1] = 0.0f;
  }
  if (w < 9)  sk[w] = kern[b * 9 + w];
  if (w == 9) sa = alpha[b];
  __syncthreads();

  const float a = sa;
  float k0[3], k1[3], k2[3];
#pragma unroll
  for (int j = 0; j < 3; ++j) { k0[j] = sk[j]; k1[j] = sk[3 + j]; k2[j] = sk[6 + j]; }

  // Rolling 3-row window of patch values and their exps.
  float p0[3], p1[3], p2[3];
  float e0[3], e1[3], e2[3];
#pragma unroll
  for (int j = 0; j < 3; ++j) {
    p0[j] = tile[0][w + j];  e0[j] = __expf(a * p0[j]);
    p1[j] = tile[1][w + j];  e1[j] = __expf(a * p1[j]);
  }

#pragma unroll
  for (int r = 0; r < TH; ++r) {
#pragma unroll
    for (int j = 0; j < 3; ++j) {             // only 3 new elements per row
      p2[j] = tile[r + 2][w + j];
      e2[j] = __expf(a * p2[j]);
    }
    float wsum = 0.0f, psum = 0.0f;
#pragma unroll
    for (int j = 0; j < 3; ++j) {
      float t;
      t = e0[j] * k0[j]; wsum += t; psum += p0[j] * t;
      t = e1[j] * k1[j]; wsum += t; psum += p1[j] * t;
      t = e2[j] * k2[j]; wsum += t; psum += p2[j] * t;
    }
    out[((size_t)b * H + (h0 + r)) * W + w] = psum / wsum;

#pragma unroll
    for (int j = 0; j < 3; ++j) {             // shift window down one row
      p0[j] = p1[j]; e0[j] = e1[j];
      p1[j] = p2[j]; e1[j] = e2[j];
    }
  }
}

// ---------------------------------------------------------------------------
extern "C" void kernel_launch(void* const* d_in, const int* in_sizes, int n_in,
                              void* d_out, int out_size, void* d_ws, size_t ws_size,
                              hipStream_t stream) {
  const float* x        = (const float*)d_in[0];  // (64,1,256,256)
  const float* features = (const float*)d_in[1];  // (64,512)
  const float* W0       = (const float*)d_in[2];  // (128,512)
  const float* b0       = (const float*)d_in[3];  // (128)
  const float* W1       = (const float*)d_in[4];  // (128,128)
  const float* b1       = (const float*)d_in[5];  // (128)
  const float* W2       = (const float*)d_in[6];  // (10,128)
  const float* b2       = (const float*)d_in[7];  // (10)
  float* out = (float*)d_out;

  float* ws    = (float*)d_ws;
  float* h0    = ws;           // 64*128 = 8192
  float* h1    = ws + 8192;    // 64*128 = 8192
  float* beta  = ws + 16384;   // 64*10  = 640
  float* alpha = ws + 17024;   // 64
  float* kern  = ws + 17088;   // 64*9   = 576

  // MLP: three WMMA GEMMs (fp32, 16x16 tile per wave)
  dim3 g01(4, 8);   // M=64 -> 4 tiles, N=128 -> 8 tiles
  gemm_wmma_relu<<<g01, 32, 0, stream>>>(features, W0, b0, h0, 64, 128, 512, 1);
  gemm_wmma_relu<<<g01, 32, 0, stream>>>(h0,       W1, b1, h1, 64, 128, 128, 1);
  dim3 g2(4, 1);    // N=10 -> 1 tile
  gemm_wmma_relu<<<g2, 32, 0, stream>>>(h1,        W2, b2, beta, 64, 10, 128, 0);

  mlp_epilogue<<<1, 64, 0, stream>>>(beta, alpha, kern);

  // Main pass: 64 images x 32 row-tiles (8 rows each), 256 threads per tile
  dim3 gs(32, 64);
  smoothmax_kernel<<<gs, 256, 0, stream>>>(x, alpha, kern, out);
}